// RetinaNetLoss_6279242187514
// MI455X (gfx1250) — compile-verified
//
#include <hip/hip_runtime.h>

// RetinaNet focal loss for (B=16, G=32, A=200000, C=1), labels all zero.
// VALU-bound problem (~1.5G lane-ops vs ~16MB HBM traffic) -> tight f32 loop,
// division-free IoU thresholding, WMMA f32 16x16x4 used as the wave32 reducer.

typedef float v2f __attribute__((ext_vector_type(2)));
typedef float v8f __attribute__((ext_vector_type(8)));

#define THREADS 256
#define NSLOT   8
#define NGT     32
#define NIMG    16

// Full 32-lane sum via V_WMMA_F32_16X16X4_F32 with B = ones:
// D[m,n] = A[m,0]+A[m,1]+A[m,2]+A[m,3] = v[m] + v[m+16]  (A vgpr1 = 0)
// lanes 0-15 hold rows 0..7 in D vgprs 0..7, lanes 16-31 hold rows 8..15.
// per-lane sum of 8 D vgprs + xor-16 shuffle = total in every lane.
__device__ __forceinline__ float wave_sum32(float v) {
  v2f a; a.x = v;    a.y = 0.0f;
  v2f b; b.x = 1.0f; b.y = 1.0f;
  v8f c = {0.f,0.f,0.f,0.f,0.f,0.f,0.f,0.f};
  c = __builtin_amdgcn_wmma_f32_16x16x4_f32(false, a, false, b, (short)0, c,
                                            false, false);
  float s = c[0] + c[1] + c[2] + c[3] + c[4] + c[5] + c[6] + c[7];
  s += __shfl_xor(s, 16, 32);
  return s;
}

// Kernel 1: 16*32 GT boxes (cx,cy,w,h,label) -> (x0,y0,x1,y1, area_g+1e-8)
__global__ void prep_gt_kernel(const float* __restrict__ y_true,
                               float* __restrict__ gtws) {
  int idx = blockIdx.x * blockDim.x + threadIdx.x;
  if (idx >= NIMG * NGT) return;
  const float* t = y_true + (size_t)idx * 5;
  float cx = t[0], cy = t[1], w = t[2], h = t[3];
  float x0 = cx - 0.5f * w, y0 = cy - 0.5f * h;
  float x1 = cx + 0.5f * w, y1 = cy + 0.5f * h;
  float* o = gtws + (size_t)idx * 5;
  o[0] = x0; o[1] = y0; o[2] = x1; o[3] = y1;
  o[4] = (x1 - x0) * (y1 - y0) + 1e-8f;   // area_g + iou eps
}

// Kernel 2: per-anchor bucket classification + focal loss, block partials.
__global__ void __launch_bounds__(THREADS)
focal_main_kernel(const float* __restrict__ anchors,
                  const float* __restrict__ probs,
                  const float* __restrict__ gtws,
                  float* __restrict__ partials,
                  int A, int stride) {
  const int b   = blockIdx.y;
  const int tid = blockIdx.x * THREADS + threadIdx.x;

  float ax0[NSLOT], ay0[NSLOT], ax1[NSLOT], ay1[NSLOT], area[NSLOT];
  float d04[NSLOT], d05[NSLOT];
  int   aidx[NSLOT];
  bool  valid[NSLOT];

#pragma unroll
  for (int s = 0; s < NSLOT; ++s) {
    int a = tid + s * stride;
    valid[s] = (a < A);
    a = valid[s] ? a : (A - 1);
    aidx[s] = a;
    float4 v = ((const float4*)anchors)[a];     // global_load_b128, coalesced
    ax0[s] = v.x; ay0[s] = v.y; ax1[s] = v.z; ay1[s] = v.w;
    area[s] = (v.z - v.x) * (v.w - v.y);
    d04[s] = -1.0f; d05[s] = -1.0f;
  }

  // GT data is uniform per block (b, g uniform) -> scalar-cache friendly.
  const float* gtb = gtws + (size_t)b * (NGT * 5);
  for (int g = 0; g < NGT; ++g) {
    float gx0 = gtb[g * 5 + 0];
    float gy0 = gtb[g * 5 + 1];
    float gx1 = gtb[g * 5 + 2];
    float gy1 = gtb[g * 5 + 3];
    float sg  = gtb[g * 5 + 4];
#pragma unroll
    for (int s = 0; s < NSLOT; ++s) {
      float lx = fmaxf(ax0[s], gx0);
      float ly = fmaxf(ay0[s], gy0);
      float rx = fminf(ax1[s], gx1);
      float ry = fminf(ay1[s], gy1);
      float w  = fmaxf(rx - lx, 0.0f);
      float h  = fmaxf(ry - ly, 0.0f);
      float inter = w * h;
      float denom = (sg + area[s]) - inter;     // area_a+area_g-inter+1e-8 > 0
      // iou >= 0.5  <=>  2*inter - denom >= 0 ; iou >= 0.4 <=> inter-0.4*denom >= 0
      d05[s] = fmaxf(d05[s], __builtin_fmaf(inter,  2.0f, -denom));
      d04[s] = fmaxf(d04[s], __builtin_fmaf(denom, -0.4f,  inter));
    }
  }

  float lsum = 0.0f, psum = 0.0f;
#pragma unroll
  for (int s = 0; s < NSLOT; ++s) {
    float p = probs[(size_t)b * A + aidx[s]];
    p = fminf(fmaxf(p, 1e-7f), 1.0f - 1e-7f);
    float omp = 1.0f - p;
    bool ge05 = d05[s] >= 0.0f;                 // target = 1
    bool ge04 = d04[s] >= 0.0f;                 // !ge04 -> target = 0
    float l1 = 0.25f * omp * omp * (-__logf(p));
    float l0 = 0.75f * p * p * (-__logf(omp));
    float loss = ge05 ? l1 : (ge04 ? 0.0f : l0);
    lsum += (valid[s] ? loss : 0.0f);
    psum += ((ge05 && valid[s]) ? 1.0f : 0.0f);
  }

  // All 256 threads active here -> EXEC all-ones for WMMA.
  float wl = wave_sum32(lsum);
  float wp = wave_sum32(psum);

  __shared__ float sl[THREADS / 32], sp[THREADS / 32];
  int wid  = threadIdx.x >> 5;
  int lane = threadIdx.x & 31;
  if (lane == 0) { sl[wid] = wl; sp[wid] = wp; }
  __syncthreads();
  if (threadIdx.x == 0) {
    float L = 0.0f, P = 0.0f;
#pragma unroll
    for (int w = 0; w < THREADS / 32; ++w) { L += sl[w]; P += sp[w]; }
    int bidx = b * gridDim.x + blockIdx.x;
    partials[bidx * 2 + 0] = L;
    partials[bidx * 2 + 1] = P;
  }
}

// Kernel 3: deterministic per-image reduce, divide by max(pos,1), mean.
__global__ void focal_final_kernel(const float* __restrict__ partials,
                                   float* __restrict__ out, int nblk) {
  __shared__ float lb[NIMG];
  int t = threadIdx.x;
  if (t < NIMG) {
    float L = 0.0f, P = 0.0f;
    for (int i = 0; i < nblk; ++i) {
      L += partials[(t * nblk + i) * 2 + 0];
      P += partials[(t * nblk + i) * 2 + 1];
    }
    lb[t] = L / fmaxf(P, 1.0f);
  }
  __syncthreads();
  if (t == 0) {
    float m = 0.0f;
#pragma unroll
    for (int i = 0; i < NIMG; ++i) m += lb[i];
    out[0] = m * (1.0f / NIMG);
  }
}

extern "C" void kernel_launch(void* const* d_in, const int* in_sizes, int n_in,
                              void* d_out, int out_size, void* d_ws, size_t ws_size,
                              hipStream_t stream) {
  const float* y_true  = (const float*)d_in[0];   // (16, 32, 5)
  const float* probs   = (const float*)d_in[1];   // (16, A, 1)
  // d_in[2] = y_regressions: unused by the reference loss -> never read (51MB saved)
  const float* anchors = (const float*)d_in[3];   // (A, 4)
  const int A = in_sizes[3] / 4;

  float* ws       = (float*)d_ws;
  float* gtws     = ws;                           // 16*32*5 = 2560 floats
  float* partials = ws + NIMG * NGT * 5;          // 2 * 16 * BX floats

  const int BX     = (A + THREADS * NSLOT - 1) / (THREADS * NSLOT);  // 98
  const int stride = BX * THREADS;

  prep_gt_kernel<<<dim3(2), 256, 0, stream>>>(y_true, gtws);
  focal_main_kernel<<<dim3(BX, NIMG), THREADS, 0, stream>>>(
      anchors, probs, gtws, partials, A, stride);
  focal_final_kernel<<<1, 32, 0, stream>>>(partials, (float*)d_out, BX);
}